// DNATransportHamiltonianGNN_17368847745106
// MI455X (gfx1250) — compile-verified
//
#include <hip/hip_runtime.h>
#include <math.h>

#define HSZ   20      // Hamiltonian size
#define NE    100     // energy grid points
#define NB    1024    // batch
#define NTRI  210     // upper-triangular element count
#define NPAD  48      // padded real-embedding size (2*20 -> 48 = 3x16 tiles)
#define LDST  52      // LDS row stride (floats), padded to dodge bank conflicts
#define WPB   4       // waves (inversions) per block

typedef __attribute__((ext_vector_type(2))) float v2f;
typedef __attribute__((ext_vector_type(8))) float v8f;

// ---- WMMA fragment loads from an LDS 16x16 tile (row stride LDST) ----------
// A operand (16x4 f32): row = lane%16 ; K = kk + 2*(lane>=16) + {0,1}
__device__ __forceinline__ v2f ldA(const float* t, int lane, int kk, float sgn) {
  int r  = lane & 15;
  int k0 = kk + ((lane >> 4) << 1);
  v2f a;
  a[0] = sgn * t[r * LDST + k0];
  a[1] = sgn * t[r * LDST + k0 + 1];
  return a;
}
// B operand (4x16 f32): col = lane%16 ; K = kk + 2*(lane>=16) + {0,1}
__device__ __forceinline__ v2f ldB(const float* t, int lane, int kk) {
  int c  = lane & 15;
  int k0 = kk + ((lane >> 4) << 1);
  v2f b;
  b[0] = t[ k0      * LDST + c];
  b[1] = t[(k0 + 1) * LDST + c];
  return b;
}
// C/D (16x16 f32): VGPR r -> row r + 8*(lane>=16), col = lane%16
__device__ __forceinline__ v8f ldC(const float* t, int lane) {
  int c = lane & 15, rb = (lane >> 4) << 3;
  v8f v;
#pragma unroll
  for (int r = 0; r < 8; ++r) v[r] = t[(rb + r) * LDST + c];
  return v;
}
__device__ __forceinline__ void stC(float* t, int lane, v8f v) {
  int c = lane & 15, rb = (lane >> 4) << 3;
#pragma unroll
  for (int r = 0; r < 8; ++r) t[(rb + r) * LDST + c] = v[r];
}

// 16x16x16 product: c += sgn * A*B, as 4 chained v_wmma_f32_16x16x4_f32
__device__ __forceinline__ v8f mm16(const float* A, const float* B, int lane,
                                    float sgn, v8f c) {
#pragma unroll
  for (int kk = 0; kk < 16; kk += 4) {
    v2f a = ldA(A, lane, kk, sgn);
    v2f b = ldB(B, lane, kk);
    c = __builtin_amdgcn_wmma_f32_16x16x4_f32(false, a, false, b, (short)0, c,
                                              false, false);
  }
  return c;
}

// In-place Gauss-Jordan inverse of a 16x16 tile in LDS (wave-cooperative,
// no pivoting: A = R + iD with D > 0 is dissipative -> generically safe).
__device__ __forceinline__ void inv16(float* T, int lane) {
  const int colh = lane & 15;          // column handled in the update phase
  const int rlo  = (lane >> 4) << 3;   // rows rlo..rlo+7
  for (int p = 0; p < 16; ++p) {
    float piv = T[p * LDST + p];
    float rp  = 1.0f / piv;
    if (lane < 16) {                   // scale pivot row; pivot entry := 1/piv
      float v = T[p * LDST + lane];
      T[p * LDST + lane] = (lane == p) ? rp : v * rp;
    }
    float prc = T[p * LDST + colh];    // scaled pivot-row value at this column
#pragma unroll
    for (int rr = 0; rr < 8; ++rr) {
      int r = rlo + rr;
      if (r != p) {
        float f = T[r * LDST + p];     // read before the col-p lane overwrites
        float v = T[r * LDST + colh];
        T[r * LDST + colh] = (colh == p) ? (-f * rp) : (v - f * prc);
      }
    }
  }
}

// ---- Stage 1: scatter upper-triangular H into the symmetric H output ------
__global__ void dna_build_H(const float* __restrict__ tri,
                            float* __restrict__ Hout) {
  int t = blockIdx.x * blockDim.x + threadIdx.x;
  if (t >= NB * HSZ * HSZ) return;
  int b  = t / (HSZ * HSZ);
  int rc = t - b * (HSZ * HSZ);
  int i = rc / HSZ, j = rc - i * HSZ;
  int lo = i < j ? i : j, hi = i < j ? j : i;
  int idx = lo * HSZ - (lo * (lo - 1)) / 2 + (hi - lo);
  Hout[t] = tri[b * NTRI + idx];
}

// ---- Stage 2: one wave32 per (batch, energy) complex inversion ------------
__global__ void __launch_bounds__(WPB * 32)
dna_negf_invert(const float* __restrict__ gL, const float* __restrict__ gR,
                const float* __restrict__ Hfull, float* __restrict__ out) {
  __shared__ float lds[WPB * NPAD * LDST];
  const int lane = threadIdx.x & 31;
  const int wave = threadIdx.x >> 5;
  float* M = &lds[wave * NPAD * LDST];

  const int inv = blockIdx.x * WPB + wave;     // grid sized exactly: no guard
  const int b = inv / NE;
  const int e = inv - b * NE;
  const float E = -3.0f + (6.0f / 99.0f) * (float)e;

  const float* Hb  = Hfull + (size_t)b * (HSZ * HSZ);
  const float* gLb = gL + b * HSZ;
  const float* gRb = gR + b * HSZ;

  // Build M = [[E*I - H, -D],[D, E*I - H]], identity-padded to 48x48.
  for (int idx = lane; idx < NPAD * NPAD; idx += 32) {
    int r = idx / NPAD, c = idx - r * NPAD;
    float val = 0.0f;
    if (r < 2 * HSZ && c < 2 * HSZ) {
      int ri = (r >= HSZ) ? r - HSZ : r;
      int ci = (c >= HSZ) ? c - HSZ : c;
      bool rIm = (r >= HSZ), cIm = (c >= HSZ);
      if (rIm == cIm) {
        val = ((ri == ci) ? E : 0.0f) - Hb[ri * HSZ + ci];
      } else if (ri == ci) {
        float d = 0.5f * (gLb[ri] + gRb[ri]);
        val = rIm ? d : -d;
      }
    } else if (r == c) {
      val = 1.0f;
    }
    M[r * LDST + c] = val;
  }

  // Blocked in-place Gauss-Jordan inversion over a 3x3 grid of 16x16 tiles.
  for (int k = 0; k < 3; ++k) {
    float* Pk = M + 16 * k * LDST + 16 * k;
    inv16(Pk, lane);                                   // Akk := inv(Akk)
    for (int j = 0; j < 3; ++j) {                      // Akj := P * Akj
      if (j == k) continue;
      float* Akj = M + 16 * k * LDST + 16 * j;
      v8f z = {};
      stC(Akj, lane, mm16(Pk, Akj, lane, 1.0f, z));
    }
    for (int i = 0; i < 3; ++i) {
      if (i == k) continue;
      float* Aik = M + 16 * i * LDST + 16 * k;
      for (int j = 0; j < 3; ++j) {                    // Aij -= Aik * Akj
        if (j == k) continue;
        float* Akj = M + 16 * k * LDST + 16 * j;
        float* Aij = M + 16 * i * LDST + 16 * j;
        v8f c = ldC(Aij, lane);
        stC(Aij, lane, mm16(Aik, Akj, lane, -1.0f, c));
      }
      v8f z = {};                                      // Aik := -Aik * P
      stC(Aik, lane, mm16(Aik, Pk, lane, -1.0f, z));
    }
  }

  // Gr = X + iY with X = M[0:20,0:20], Y = M[20:40,0:20].
  float accT = 0.0f;
  for (int idx = lane; idx < HSZ * HSZ; idx += 32) {
    int i = idx / HSZ, j = idx - i * HSZ;
    float x = M[i * LDST + j];
    float y = M[(HSZ + i) * LDST + j];
    accT += gLb[i] * gRb[j] * (x * x + y * y);
  }
  float accD = (lane < HSZ) ? M[(HSZ + lane) * LDST + lane] : 0.0f;
#pragma unroll
  for (int off = 16; off > 0; off >>= 1) {
    accT += __shfl_xor(accT, off, 32);
    accD += __shfl_xor(accD, off, 32);
  }
  if (lane == 0) {
    out[(size_t)b * NE + e] = log10f(fmaxf(accT, 1e-16f));                 // T_log
    out[(size_t)NB * NE + (size_t)b * NE + e] = log10f(fmaxf(-accD, 1e-16f)); // DOS_log
  }
}

extern "C" void kernel_launch(void* const* d_in, const int* in_sizes, int n_in,
                              void* d_out, int out_size, void* d_ws, size_t ws_size,
                              hipStream_t stream) {
  (void)in_sizes; (void)n_in; (void)out_size; (void)d_ws; (void)ws_size;
  const float* tri = (const float*)d_in[0];   // [B, 210]
  const float* gL  = (const float*)d_in[1];   // [B, 20]
  const float* gR  = (const float*)d_in[2];   // [B, 20]
  float* out  = (float*)d_out;                // T_log | DOS_log | H
  float* outH = out + (size_t)2 * NB * NE;    // H block (also stage-2 input)

  int nH = NB * HSZ * HSZ;
  dna_build_H<<<(nH + 255) / 256, 256, 0, stream>>>(tri, outH);

  int blocks = (NB * NE) / WPB;               // 102400 inversions / 4 waves
  dna_negf_invert<<<blocks, WPB * 32, 0, stream>>>(gL, gR, outH, out);
}